// ResidualBlock_48318382080047
// MI455X (gfx1250) — compile-verified
//
#include <hip/hip_runtime.h>
#include <hip/hip_bf16.h>
#include <math.h>

// ---------------------------------------------------------------------------
// Shapes (fixed by the reference)
// ---------------------------------------------------------------------------
#define Bsz 4
#define Cch 128
#define Hh  64
#define Ww  64
#define Ll  (Hh * Ww)        // 4096
#define Rows (Bsz * Ll)      // 16384
#define Ee  256
#define Nn  16
#define Rr  8
#define Kk  4

typedef __attribute__((ext_vector_type(2))) float v2f;
typedef __attribute__((ext_vector_type(8))) float v8f;
typedef __attribute__((ext_vector_type(4))) unsigned int u32x4;
typedef __attribute__((ext_vector_type(8))) int i32x8;
typedef __attribute__((ext_vector_type(4))) int i32x4;

#if __has_builtin(__builtin_amdgcn_tensor_load_to_lds) && __has_builtin(__builtin_amdgcn_s_wait_tensorcnt)
#define HAVE_TDM 1
#else
#define HAVE_TDM 0
#endif

// ---------------------------------------------------------------------------
// Transpose x (B,C,L) -> (B*L, C)
// ---------------------------------------------------------------------------
__global__ __launch_bounds__(256)
void transpose_in_kernel(const float* __restrict__ x, float* __restrict__ xo) {
    int idx = blockIdx.x * 256 + threadIdx.x;       // over B*L*C
    int c = idx & (Cch - 1);
    int t = idx >> 7;                                // b*L + l
    int l = t & (Ll - 1);
    int b = t >> 12;
    xo[idx] = x[((size_t)b * Cch + c) * Ll + l];
}

// ---------------------------------------------------------------------------
// BatchNorm stats: deterministic two-pass reduction over 16384 rows x 128 ch
// ---------------------------------------------------------------------------
__global__ __launch_bounds__(128)
void bn_partial_kernel(const float* __restrict__ xin, float* __restrict__ part) {
    int c = threadIdx.x;                 // channel
    int blk = blockIdx.x;                // 128 blocks x 128 rows
    size_t base = (size_t)blk * 128 * Cch;
    float s = 0.f, ss = 0.f;
    for (int r = 0; r < 128; ++r) {
        float x = xin[base + (size_t)r * Cch + c];
        s += x; ss += x * x;
    }
    part[blk * 256 + c] = s;
    part[blk * 256 + 128 + c] = ss;
}

__global__ __launch_bounds__(128)
void bn_final_kernel(const float* __restrict__ part, const float* __restrict__ g,
                     const float* __restrict__ bt, float* __restrict__ coef) {
    int c = threadIdx.x;
    float s = 0.f, ss = 0.f;
    for (int b = 0; b < 128; ++b) { s += part[b * 256 + c]; ss += part[b * 256 + 128 + c]; }
    float mean = s * (1.f / (float)Rows);
    float var  = ss * (1.f / (float)Rows) - mean * mean;
    float a = g[c] * rsqrtf(var + 1e-5f);
    coef[c] = a;
    coef[128 + c] = bt[c] - mean * a;
}

// ---------------------------------------------------------------------------
// BN affine + leaky-relu + LayerNorm over C (one wave32 per row, 4 ch/lane)
// ---------------------------------------------------------------------------
__global__ __launch_bounds__(256)
void norm_ln_kernel(const float* __restrict__ xin, const float* __restrict__ coef,
                    const float* __restrict__ lg, const float* __restrict__ lb,
                    float* __restrict__ u, float* __restrict__ un) {
    int lane = threadIdx.x & 31;
    int wave = threadIdx.x >> 5;
    int row  = blockIdx.x * 8 + wave;
    size_t base = (size_t)row * Cch;
    float v[4];
    float s = 0.f;
    #pragma unroll
    for (int j = 0; j < 4; ++j) {
        int c = lane + 32 * j;
        float x = xin[base + c];
        float t = coef[c] * x + coef[128 + c];
        t = (t > 0.f) ? t : 0.01f * t;        // leaky relu
        v[j] = t;
        u[base + c] = t;
        s += t;
    }
    #pragma unroll
    for (int o = 16; o >= 1; o >>= 1) s += __shfl_xor(s, o, 32);
    float mean = s * (1.f / (float)Cch);
    float ss = 0.f;
    #pragma unroll
    for (int j = 0; j < 4; ++j) { float d = v[j] - mean; ss += d * d; }
    #pragma unroll
    for (int o = 16; o >= 1; o >>= 1) ss += __shfl_xor(ss, o, 32);
    float rs = rsqrtf(ss * (1.f / (float)Cch) + 1e-5f);
    #pragma unroll
    for (int j = 0; j < 4; ++j) {
        int c = lane + 32 * j;
        un[base + c] = (v[j] - mean) * rs * lg[c] + lb[c];
    }
}

// ---------------------------------------------------------------------------
// Generic f32 WMMA GEMM with TDM-staged B panel in LDS.
// D[M x N] = epilogue(A[M x K] @ B[K x N])
// Wave -> 16 x (NT*16) tile, 4 waves/block -> 64 rows/block.
// B staged in K-chunks of 128 rows via tensor_load_to_lds with LDS padding
// (pitch = TW+8 dwords) so hi/lo half-wave ds reads hit disjoint banks.
// EPI: 0 = store raw, 1 = softplus(v + bias[col]), 2 = v + addend[row,col]
// ---------------------------------------------------------------------------
#define KCHUNK 128

template<int KSIZE, int NT, int EPI>
__global__ __launch_bounds__(128)
void wmma_gemm_kernel(const float* __restrict__ A, int lda,
                      const float* __restrict__ B, int ldb,
                      float* __restrict__ D, int ldd,
                      const float* __restrict__ bias,
                      const float* __restrict__ addend) {
    constexpr int TW    = NT * 16;        // tile width (cols)
    constexpr int PITCH = TW + 8;         // LDS row pitch (TDM pad: 8 dwords)
    constexpr int PADI  = (TW == 64) ? 5 : 4;   // pad every 64 / 32 dwords
    __shared__ float btile[KCHUNK * PITCH];

    const int lane = threadIdx.x & 31;
    const int wave = threadIdx.x >> 5;
    const int m    = lane & 15;
    const int hi   = lane >> 4;
    const int row0 = (blockIdx.y * 4 + wave) * 16;
    const int col0 = blockIdx.x * TW;

    v8f acc[NT] = {};
    const float* Arow = A + (size_t)(row0 + m) * lda;

    for (int kc = 0; kc < KSIZE; kc += KCHUNK) {
        const float* Bc = B + (size_t)kc * ldb + col0;
#if HAVE_TDM
        if (threadIdx.x < 32) {
            // Build Tensor DMA Descriptor (D#): 2-D tile, 4-byte elements,
            // tile = KCHUNK x TW, row stride = ldb, LDS pad 8 dwords / row.
            unsigned long long gaddr = (unsigned long long)(const void*)Bc;
            const unsigned td0 = 0x10000000u, td1 = 0x10000000u; // huge dims: no OOB clip
            u32x4 g0;
            g0.x = 1u;                                            // count=1
            g0.y = 0u;                                            // lds_addr = 0 (btile)
            g0.z = (unsigned)(gaddr & 0xffffffffull);
            g0.w = (unsigned)((gaddr >> 32) & 0x1ffffffull) | (2u << 30); // type=2
            i32x8 g1;
            g1[0] = (int)((2u << 16) | (1u << 20) |               // data_size=4B, pad_enable
                          ((unsigned)PADI << 22) | (7u << 25));   // pad_interval, pad=8 dwords
            g1[1] = (int)((td0 & 0xffffu) << 16);                 // tensor_dim0[15:0]
            g1[2] = (int)((td0 >> 16) | ((td1 & 0xffffu) << 16)); // td0[31:16] | td1[15:0]
            g1[3] = (int)((td1 >> 16) | ((unsigned)TW << 16));    // td1[31:16] | tile_dim0
            g1[4] = (int)KCHUNK;                                  // tile_dim1 (tile_dim2=0)
            g1[5] = (int)(unsigned)ldb;                           // tensor_dim0_stride[31:0]
            g1[6] = 0;                                            // stride hi / dim1 stride
            g1[7] = 0;
            i32x4 gz4 = {0, 0, 0, 0};
            i32x8 gz8 = {0, 0, 0, 0, 0, 0, 0, 0};
            __builtin_amdgcn_tensor_load_to_lds(g0, g1, gz4, gz4, gz8, 0);
            __builtin_amdgcn_s_wait_tensorcnt(0);
        }
#else
        for (int idx = threadIdx.x; idx < KCHUNK * TW; idx += 128) {
            int r  = idx / TW;
            int cx = idx - r * TW;
            btile[r * PITCH + cx] = Bc[(size_t)r * ldb + cx];
        }
#endif
        __syncthreads();

        const float* Achunk = Arow + kc;
        for (int k = 0; k < KCHUNK; k += 4) {
            const int ka = k + 2 * hi;
            v2f a = *(const v2f*)(Achunk + ka);   // 8B-aligned (lda, ka even)
            #pragma unroll
            for (int t = 0; t < NT; ++t) {
                v2f b;
                b[0] = btile[ka * PITCH + t * 16 + m];
                b[1] = btile[(ka + 1) * PITCH + t * 16 + m];
                acc[t] = __builtin_amdgcn_wmma_f32_16x16x4_f32(
                    false, a, false, b, (short)0, acc[t], false, false);
            }
        }
        __syncthreads();
    }

    #pragma unroll
    for (int t = 0; t < NT; ++t) {
        int col = col0 + t * 16 + m;
        float bcol = (EPI == 1) ? bias[col] : 0.f;
        #pragma unroll
        for (int j = 0; j < 8; ++j) {
            int row = row0 + j + hi * 8;
            float v = acc[t][j];
            if constexpr (EPI == 1) {          // softplus(v + b_dt)
                v += bcol;
                v = (v > 20.f) ? v : log1pf(expf(v));
            } else if constexpr (EPI == 2) {   // residual add (u + out)
                v += addend[(size_t)row * ldd + col];
            }
            D[(size_t)row * ldd + col] = v;
        }
    }
}

// ---------------------------------------------------------------------------
// Depthwise causal conv (K=4) along L + bias + SiLU.  xz: rows x 512, xm: rows x 256
// ---------------------------------------------------------------------------
__global__ __launch_bounds__(256)
void conv_silu_kernel(const float* __restrict__ xz, const float* __restrict__ w,
                      const float* __restrict__ bconv, float* __restrict__ xm) {
    int idx = blockIdx.x * 256 + threadIdx.x;    // row*256 + e
    int e   = idx & (Ee - 1);
    int row = idx >> 8;
    int l   = row & (Ll - 1);
    const float* xc = xz + (size_t)row * (2 * Ee) + e;
    float w0 = w[e * Kk + 0], w1 = w[e * Kk + 1], w2 = w[e * Kk + 2], w3 = w[e * Kk + 3];
    float acc = w3 * xc[0];
    if (l >= 1) acc = fmaf(w2, xc[-(2 * Ee)], acc);
    if (l >= 2) acc = fmaf(w1, xc[-(4 * Ee)], acc);
    if (l >= 3) acc = fmaf(w0, xc[-(6 * Ee)], acc);
    acc += bconv[e];
    xm[idx] = acc / (1.f + expf(-acc));          // silu
}

// ---------------------------------------------------------------------------
// Fold W_xp[:, :8] @ W_dt  ->  Wcomb (256 x 256)
// ---------------------------------------------------------------------------
__global__ __launch_bounds__(256)
void wcomb_kernel(const float* __restrict__ Wxp, const float* __restrict__ Wdt,
                  float* __restrict__ Wc) {
    int e = threadIdx.x;
    int k = blockIdx.x;
    float s = 0.f;
    #pragma unroll
    for (int r = 0; r < Rr; ++r) s = fmaf(Wxp[k * (Rr + 2 * Nn) + r], Wdt[r * Ee + e], s);
    Wc[k * Ee + e] = s;
}

// ---------------------------------------------------------------------------
// Selective scan: 1024 independent (b,e) recurrences; 16 lanes = 16 states n.
// Read streams are prefetched 24 timesteps ahead (global_prefetch_b8) so the
// 4096-step recurrence only waits on the exp/FMA chain.
// Fused epilogue: y = (scan + Dp*xm) * silu(z)
// ---------------------------------------------------------------------------
__global__ __launch_bounds__(256)
void scan_kernel(const float* __restrict__ dt, const float* __restrict__ BC,
                 const float* __restrict__ xm, const float* __restrict__ xz,
                 const float* __restrict__ A_log, const float* __restrict__ Dp,
                 float* __restrict__ y) {
    int g   = blockIdx.x * 256 + threadIdx.x;
    int n   = g & (Nn - 1);
    int grp = g >> 4;                 // 0..1023 == b*256 + e
    int e   = grp & (Ee - 1);
    int b   = grp >> 8;
    float An  = -expf(A_log[e * Nn + n]);
    float Dpe = Dp[e];
    float h = 0.f;
    size_t rbase = (size_t)b * Ll;
    for (int l = 0; l < Ll; ++l) {
        size_t row = rbase + l;
        size_t prow = row + 24;        // prefetch distance (stays inside ws)
        __builtin_prefetch(dt + prow * Ee + e, 0, 1);
        __builtin_prefetch(xm + prow * Ee + e, 0, 1);
        __builtin_prefetch(BC + prow * (2 * Nn) + n, 0, 1);
        __builtin_prefetch(xz + prow * (2 * Ee) + Ee + e, 0, 1);
        float dtv = dt[row * Ee + e];
        float xv  = xm[row * Ee + e];
        float Bv  = BC[row * (2 * Nn) + n];
        float Cv  = BC[row * (2 * Nn) + Nn + n];
        h = expf(dtv * An) * h + dtv * Bv * xv;
        float p = h * Cv;
        p += __shfl_xor(p, 1, 16);
        p += __shfl_xor(p, 2, 16);
        p += __shfl_xor(p, 4, 16);
        p += __shfl_xor(p, 8, 16);
        if (n == 0) {
            float zv = xz[row * (2 * Ee) + Ee + e];
            float yv = (p + Dpe * xv) * (zv / (1.f + expf(-zv)));
            y[row * Ee + e] = yv;
        }
    }
}

// ---------------------------------------------------------------------------
// Final: out(B,C,L) = x(B,C,L) + stage2_out(B*L, C)
// ---------------------------------------------------------------------------
__global__ __launch_bounds__(256)
void final_add_kernel(const float* __restrict__ s2, const float* __restrict__ x,
                      float* __restrict__ out) {
    int idx = blockIdx.x * 256 + threadIdx.x;    // over B*C*L (output layout)
    int l = idx & (Ll - 1);
    int t = idx >> 12;                            // b*C + c
    int c = t & (Cch - 1);
    int b = t >> 7;
    out[idx] = x[idx] + s2[((size_t)b * Ll + l) * Cch + c];
}

// ---------------------------------------------------------------------------
// Host-side stage driver
// ---------------------------------------------------------------------------
struct Params {
    const float *bn_g, *bn_b, *ln_g, *ln_b, *W_in, *conv_w, *conv_b;
    const float *W_xp, *W_dt, *b_dt, *A_log, *Dp, *W_out;
};

static void run_stage(int i, const float* xin, float* out, const Params& p,
                      float* u, float* un, float* xz, float* xm, float* BC,
                      float* dt, float* y, float* part, float* coef, float* wcomb,
                      hipStream_t stream) {
    const float* bn_g  = p.bn_g  + i * Cch;
    const float* bn_b  = p.bn_b  + i * Cch;
    const float* ln_g  = p.ln_g  + i * Cch;
    const float* ln_b  = p.ln_b  + i * Cch;
    const float* W_in  = p.W_in  + (size_t)i * Cch * 2 * Ee;
    const float* cw    = p.conv_w + (size_t)i * Ee * Kk;
    const float* cb    = p.conv_b + (size_t)i * Ee;
    const float* W_xp  = p.W_xp  + (size_t)i * Ee * (Rr + 2 * Nn);
    const float* W_dt  = p.W_dt  + (size_t)i * Rr * Ee;
    const float* b_dt  = p.b_dt  + (size_t)i * Ee;
    const float* A_log = p.A_log + (size_t)i * Ee * Nn;
    const float* Dp    = p.Dp    + (size_t)i * Ee;
    const float* W_out = p.W_out + (size_t)i * Ee * Cch;

    bn_partial_kernel<<<128, 128, 0, stream>>>(xin, part);
    bn_final_kernel<<<1, 128, 0, stream>>>(part, bn_g, bn_b, coef);
    norm_ln_kernel<<<Rows / 8, 256, 0, stream>>>(xin, coef, ln_g, ln_b, u, un);

    // xz = un @ W_in   (16384x128 @ 128x512)
    wmma_gemm_kernel<Cch, 4, 0><<<dim3((2 * Ee) / 64, Rows / 64), 128, 0, stream>>>(
        un, Cch, W_in, 2 * Ee, xz, 2 * Ee, nullptr, nullptr);

    // xm = silu(depthwise_conv(xz[:, :256]) + conv_b)
    conv_silu_kernel<<<(Rows * Ee) / 256, 256, 0, stream>>>(xz, cw, cb, xm);

    // Wcomb = W_xp[:, :8] @ W_dt
    wcomb_kernel<<<Ee, Ee, 0, stream>>>(W_xp, W_dt, wcomb);

    // BC = xm @ W_xp[:, 8:40]   (16384x256 @ 256x32)
    wmma_gemm_kernel<Ee, 2, 0><<<dim3(1, Rows / 64), 128, 0, stream>>>(
        xm, Ee, W_xp + Rr, Rr + 2 * Nn, BC, 2 * Nn, nullptr, nullptr);

    // dt = softplus(xm @ Wcomb + b_dt)   (16384x256 @ 256x256)
    wmma_gemm_kernel<Ee, 4, 1><<<dim3(Ee / 64, Rows / 64), 128, 0, stream>>>(
        xm, Ee, wcomb, Ee, dt, Ee, b_dt, nullptr);

    // sequential selective scan + gating
    scan_kernel<<<(Bsz * Ee * Nn) / 256, 256, 0, stream>>>(dt, BC, xm, xz, A_log, Dp, y);

    // out = u + y @ W_out   (16384x256 @ 256x128)
    wmma_gemm_kernel<Ee, 4, 2><<<dim3(Cch / 64, Rows / 64), 128, 0, stream>>>(
        y, Ee, W_out, Cch, out, Cch, nullptr, u);
}

extern "C" void kernel_launch(void* const* d_in, const int* in_sizes, int n_in,
                              void* d_out, int out_size, void* d_ws, size_t ws_size,
                              hipStream_t stream) {
    (void)in_sizes; (void)n_in; (void)out_size; (void)ws_size;

    const float* x = (const float*)d_in[0];
    Params p;
    p.bn_g   = (const float*)d_in[1];
    p.bn_b   = (const float*)d_in[2];
    p.ln_g   = (const float*)d_in[3];
    p.ln_b   = (const float*)d_in[4];
    p.W_in   = (const float*)d_in[5];
    p.conv_w = (const float*)d_in[6];
    p.conv_b = (const float*)d_in[7];
    p.W_xp   = (const float*)d_in[8];
    p.W_dt   = (const float*)d_in[9];
    p.b_dt   = (const float*)d_in[10];
    p.A_log  = (const float*)d_in[11];
    p.Dp     = (const float*)d_in[12];
    p.W_out  = (const float*)d_in[13];

    float* ws = (float*)d_ws;
    size_t o = 0;
    float* xin0  = ws + o; o += (size_t)Rows * Cch;        // 2,097,152
    float* u     = ws + o; o += (size_t)Rows * Cch;
    float* un    = ws + o; o += (size_t)Rows * Cch;
    float* xz    = ws + o; o += (size_t)Rows * 2 * Ee;     // 8,388,608
    float* xm    = ws + o; o += (size_t)Rows * Ee;
    float* BC    = ws + o; o += (size_t)Rows * 2 * Nn;
    float* dt    = ws + o; o += (size_t)Rows * Ee;
    float* y     = ws + o; o += (size_t)Rows * Ee;
    float* s1out = ws + o; o += (size_t)Rows * Cch;
    float* part  = ws + o; o += 128 * 256;
    float* coef  = ws + o; o += 256;
    float* wcomb = ws + o; o += (size_t)Ee * Ee;

    // x (B,C,L) -> xin0 (B*L, C)
    transpose_in_kernel<<<(Rows * Cch) / 256, 256, 0, stream>>>(x, xin0);

    // stage 0: xin0 -> s1out ; stage 1: s1out -> xin0 (xin0 is dead by then)
    run_stage(0, xin0, s1out, p, u, un, xz, xm, BC, dt, y, part, coef, wcomb, stream);
    run_stage(1, s1out, xin0, p, u, un, xz, xm, BC, dt, y, part, coef, wcomb, stream);

    // out = x + stage2_out  (back to B,C,H,W layout)
    final_add_kernel<<<(Rows * Cch) / 256, 256, 0, stream>>>(xin0, x, (float*)d_out);
}